// MMEAttn_65532611003048
// MI455X (gfx1250) — compile-verified
//
#include <hip/hip_runtime.h>
#include <stdint.h>

// ---- problem constants (from reference) ----
#define BATCH 64
#define KVT   256
#define NHEAD 32
#define HDIM  128
#define MH    (NHEAD * HDIM)   // 4096

// LDS K-tile row stride (floats): 128 + 4 pad -> bank-conflict-free
// for the ds_load_b64 A-tile read pattern (banks 4i+2g+{0,1} all distinct).
#define KSTRIDE 132

typedef __attribute__((ext_vector_type(2))) float v2f;
typedef __attribute__((ext_vector_type(4))) float v4f;
typedef __attribute__((ext_vector_type(8))) float v8f;

// Exact pointee type from the builtin's signature (per clang diagnostic):
// int __attribute__((vector_size(16))), in AS1 (global) / AS3 (LDS).
typedef int v4i_vs __attribute__((vector_size(16)));
typedef __attribute__((address_space(1))) v4i_vs as1_v4i;
typedef __attribute__((address_space(3))) v4i_vs as3_v4i;

#ifndef __has_builtin
#define __has_builtin(x) 0
#endif

// Async global->LDS copy of 16B per lane (GLOBAL_LOAD_ASYNC_TO_LDS_B128,
// tracked by ASYNCcnt; no data VGPRs consumed).
__device__ __forceinline__ void async_load_b128(const float* g, float* l)
{
#if __has_builtin(__builtin_amdgcn_global_load_async_to_lds_b128)
    __builtin_amdgcn_global_load_async_to_lds_b128(
        (as1_v4i*)(uintptr_t)g,
        (as3_v4i*)(uint32_t)(uintptr_t)l,   // low 32 bits = LDS byte offset
        0, 0);
#else
    uint32_t loff  = (uint32_t)(uintptr_t)l;
    uint64_t gaddr = (uint64_t)(uintptr_t)g;
    asm volatile("global_load_async_to_lds_b128 %0, %1, off"
                 :: "v"(loff), "v"(gaddr) : "memory");
#endif
}

template <int N>
__device__ __forceinline__ void wait_asynccnt()
{
#if __has_builtin(__builtin_amdgcn_s_wait_asynccnt)
    __builtin_amdgcn_s_wait_asynccnt((unsigned short)N);
#else
    asm volatile("s_wait_asynccnt %0" :: "n"(N) : "memory");
#endif
    asm volatile("" ::: "memory");   // keep LDS reads below the wait
}

__global__ __launch_bounds__(32) void mme_attn_kernel(
    const float* __restrict__ qg_,
    const float* __restrict__ kg_,
    const float* __restrict__ vg_,
    float* __restrict__ outg)
{
    __shared__ float q_lds[HDIM];
    __shared__ float k_lds[2][16 * KSTRIDE];   // double-buffered K tile
    __shared__ float s_lds[KVT];

    const int lane = threadIdx.x;          // 0..31 (wave32)
    const int pair = blockIdx.x;           // 0..2047
    const int b    = pair >> 5;
    const int m    = pair & (NHEAD - 1);

    const float* qg = qg_ + (size_t)b * MH + (size_t)m * HDIM;
    const float* kg = kg_ + (size_t)b * KVT * MH + (size_t)m * HDIM;
    const float* vg = vg_ + (size_t)b * KVT * MH + (size_t)m * HDIM;

    // ---- q (128 floats) -> LDS: one b128 per lane ----
    *(v4f*)(&q_lds[4 * lane]) = *(const v4f*)(qg + 4 * lane);

    const int g = lane >> 4;    // lane group (0 or 1) in WMMA layout
    const int i = lane & 15;    // position within group

    // 16 K rows -> LDS buffer `buf` (one async b128 per lane per row)
    auto issue_tile = [&](int t0, int buf) {
        asm volatile("" ::: "memory");     // don't hoist above prior LDS reads
        #pragma unroll
        for (int r = 0; r < 16; ++r)
            async_load_b128(kg + (size_t)(t0 + r) * MH + 4 * lane,
                            &k_lds[buf][r * KSTRIDE + 4 * lane]);
    };

    // ---- stage 1: scores via V_WMMA_F32_16X16X4_F32, double-buffered ----
    // score[t] = sum_h K[t][h]*q[h]; 16 kv rows per D tile (M dim),
    // q broadcast across all 16 N columns, contraction over h in steps of 4.
    issue_tile(0, 0);

    #pragma unroll 1
    for (int t0 = 0; t0 < KVT; t0 += 16) {
        const int buf = (t0 >> 4) & 1;
        if (t0 + 16 < KVT) {
            issue_tile(t0 + 16, buf ^ 1);  // prefetch next tile
            wait_asynccnt<16>();           // in-order: current tile complete
        } else {
            wait_asynccnt<0>();
        }

        v8f acc = {};
        #pragma unroll
        for (int h0 = 0; h0 < HDIM; h0 += 4) {
            // A tile: lane(g,i), vgpr j -> A[M=i][K=2g+j] = K[t0+i][h0+2g+j]
            v2f a  = *(const v2f*)(&k_lds[buf][i * KSTRIDE + h0 + 2 * g]);
            // B tile: lane(g,i), vgpr j -> B[K=2g+j][N=i] = q[h0+2g+j]
            v2f bq = *(const v2f*)(&q_lds[h0 + 2 * g]);
            acc = __builtin_amdgcn_wmma_f32_16x16x4_f32(
                false, a, false, bq, (short)0, acc, false, false);
        }

        // D layout: lane(g,i), vgpr r -> M = r + 8g, N = i. Column N=0 has
        // the 16 scores; lanes 0 and 16 write them to LDS.
        if (i == 0) {
            #pragma unroll
            for (int r = 0; r < 8; ++r)
                s_lds[t0 + 8 * g + r] = acc[r];
        }
    }
    asm volatile("s_wait_dscnt 0" ::: "memory");

    // ---- softmax over 256 scores (wave32 reductions) ----
    float sv[8];
    float mx = -3.402823466e+38f;
    #pragma unroll
    for (int j = 0; j < 8; ++j) {
        sv[j] = s_lds[lane + 32 * j];
        mx = fmaxf(mx, sv[j]);
    }
    #pragma unroll
    for (int off = 16; off > 0; off >>= 1)
        mx = fmaxf(mx, __shfl_xor(mx, off, 32));

    float sum = 0.0f;
    #pragma unroll
    for (int j = 0; j < 8; ++j) {
        sv[j] = __expf(sv[j] - mx);
        sum += sv[j];
    }
    #pragma unroll
    for (int off = 16; off > 0; off >>= 1)
        sum += __shfl_xor(sum, off, 32);

    const float inv = 1.0f / sum;
    #pragma unroll
    for (int j = 0; j < 8; ++j)
        s_lds[lane + 32 * j] = sv[j] * inv;
    asm volatile("s_wait_dscnt 0" ::: "memory");

    // ---- stage 2: out[h] = sum_t attn[t] * V[t][h] ----
    // Pure HBM stream (strictly read-once -> nontemporal): each lane owns
    // 4 h's; one b128 per lane per row, attn[t] broadcast from LDS.
    v4f acc2 = {};
    #pragma unroll 8
    for (int t = 0; t < KVT; ++t) {
        float w = s_lds[t];                                        // ds broadcast
        const v4f* vp = (const v4f*)(vg + (size_t)t * MH + 4 * lane);
        v4f vv = __builtin_nontemporal_load(vp);                   // b128, TH=NT
        acc2 += vv * w;
    }

    // out layout [B, M, 1, H] -> flat pair*128 + h
    __builtin_nontemporal_store(acc2, (v4f*)(outg + (size_t)pair * HDIM + 4 * lane));
}

extern "C" void kernel_launch(void* const* d_in, const int* in_sizes, int n_in,
                              void* d_out, int out_size, void* d_ws, size_t ws_size,
                              hipStream_t stream)
{
    const float* q = (const float*)d_in[0];
    const float* k = (const float*)d_in[1];
    const float* v = (const float*)d_in[2];
    float* out = (float*)d_out;

    dim3 grid(BATCH * NHEAD);  // 2048 single-wave workgroups
    dim3 block(32);
    mme_attn_kernel<<<grid, block, 0, stream>>>(q, k, v, out);
}